// GNNLayer_83923660964215
// MI455X (gfx1250) — compile-verified
//
#include <hip/hip_runtime.h>

#define DIM 128
#define LN_EPS 1e-5f
#define LDSTRIDE 132   // 16x128 xn tile in LDS, padded: bank = (4*m + k) % 64, conflict-free

typedef float v2f __attribute__((ext_vector_type(2)));
typedef float v8f __attribute__((ext_vector_type(8)));

// ---------------- degree / normalization ----------------
__global__ void init_deg_kernel(float* __restrict__ deg, int n) {
    int i = blockIdx.x * blockDim.x + threadIdx.x;
    if (i < n) deg[i] = 1.0f;                     // self-loop weight
}

__global__ void edge_deg_kernel(const int* __restrict__ ei, const float* __restrict__ ew,
                                float* __restrict__ deg, int E) {
    int e = blockIdx.x * blockDim.x + threadIdx.x;
    if (e < E) atomicAdd(&deg[ei[E + e]], ew[e]); // col = target
}

__global__ void dinv_kernel(const float* __restrict__ deg, float* __restrict__ dinv, int n) {
    int i = blockIdx.x * blockDim.x + threadIdx.x;
    if (i < n) {
        float d = deg[i];
        dinv[i] = d > 0.0f ? rsqrtf(d) : 0.0f;
    }
}

// ---------------- LayerNorm + GEMM (h = LN(x) @ W^T) via WMMA f32 ----------------
// block = 128 threads (4 waves), 16 nodes per block, full K=128, full Nout=128.
__global__ __launch_bounds__(128) void ln_gemm_wmma_kernel(
        const float* __restrict__ x, const float* __restrict__ W,
        const float* __restrict__ lnw, const float* __restrict__ lnb,
        float* __restrict__ h, int n) {
    __shared__ float xn[16 * LDSTRIDE];

    const int tid   = threadIdx.x;
    const int node  = tid >> 3;          // 0..15 (local node)
    const int grp   = tid & 7;           // 8 lanes cooperate per node
    const int gbase = blockIdx.x * 16;

    int gn = gbase + node;
    if (gn >= n) gn = n - 1;             // clamp (N=50000 is a multiple of 16 anyway)
    const float* xp = x + (size_t)gn * DIM;

    // ----- Phase 1: LayerNorm (per-node mean/var over 128 feats) -----
    float4 v[4];
    float s = 0.0f, s2 = 0.0f;
#pragma unroll
    for (int i = 0; i < 4; ++i) {
        int f = grp * 4 + i * 32;        // strided so each wave iter is contiguous
        v[i] = *(const float4*)(xp + f);
        s  += v[i].x + v[i].y + v[i].z + v[i].w;
        s2 += v[i].x * v[i].x + v[i].y * v[i].y + v[i].z * v[i].z + v[i].w * v[i].w;
    }
    // reduce across the 8 lanes owning this node (lane bits 0..2)
#pragma unroll
    for (int m = 1; m <= 4; m <<= 1) {
        s  += __shfl_xor(s,  m, 32);
        s2 += __shfl_xor(s2, m, 32);
    }
    const float mu   = s * (1.0f / DIM);
    const float var  = s2 * (1.0f / DIM) - mu * mu;
    const float rstd = rsqrtf(var + LN_EPS);

#pragma unroll
    for (int i = 0; i < 4; ++i) {
        int f = grp * 4 + i * 32;
        float4 w4 = *(const float4*)(lnw + f);
        float4 b4 = *(const float4*)(lnb + f);
        float* dst = &xn[node * LDSTRIDE + f];
        dst[0] = (v[i].x - mu) * rstd * w4.x + b4.x;
        dst[1] = (v[i].y - mu) * rstd * w4.y + b4.y;
        dst[2] = (v[i].z - mu) * rstd * w4.z + b4.z;
        dst[3] = (v[i].w - mu) * rstd * w4.w + b4.w;
    }
    __syncthreads();

    // ----- Phase 2: 16x128 @ 128x128 GEMM, V_WMMA_F32_16X16X4_F32 -----
    // wave w computes output column tiles [32w, 32w+16) and [32w+16, 32w+32)
    const int wave  = tid >> 5;
    const int lane  = tid & 31;
    const int m16   = lane & 15;
    const int koff  = (lane >> 4) << 1;        // 0 for lanes 0-15, 2 for lanes 16-31 (ISA A layout)
    const int nbase = wave * 32;

    const float* w0   = W + (size_t)(nbase +      m16) * DIM + koff;  // B[k][n] = W[n][k]
    const float* w1   = W + (size_t)(nbase + 16 + m16) * DIM + koff;
    const float* arow = &xn[m16 * LDSTRIDE + koff];

    v8f acc0 = {0.f, 0.f, 0.f, 0.f, 0.f, 0.f, 0.f, 0.f};
    v8f acc1 = {0.f, 0.f, 0.f, 0.f, 0.f, 0.f, 0.f, 0.f};

#pragma unroll
    for (int k0 = 0; k0 < DIM; k0 += 4) {
        v2f a  = *(const v2f*)(arow + k0);     // LDS, ds_load_b64
        v2f b0 = *(const v2f*)(w0 + k0);       // global, cached (W = 64KB)
        v2f b1 = *(const v2f*)(w1 + k0);
        acc0 = __builtin_amdgcn_wmma_f32_16x16x4_f32(false, a, false, b0,
                                                     (short)0, acc0, false, false);
        acc1 = __builtin_amdgcn_wmma_f32_16x16x4_f32(false, a, false, b1,
                                                     (short)0, acc1, false, false);
    }

    // C/D layout: VGPR r -> M=r (lanes 0-15), M=8+r (lanes 16-31); N = lane&15
    const int nlo = lane & 15;
    const int mhi = (lane >> 4) << 3;
#pragma unroll
    for (int r = 0; r < 8; ++r) {
        int gm = gbase + mhi + r;
        if (gm < n) {
            h[(size_t)gm * DIM + nbase +      nlo] = acc0[r];
            h[(size_t)gm * DIM + nbase + 16 + nlo] = acc1[r];
        }
    }
}

// ---------------- self-loop term initializes d_out (norm = dinv[i]^2) ----------------
__global__ void selfloop_init_kernel(const float* __restrict__ h, const float* __restrict__ dinv,
                                     float* __restrict__ out, int total) {
    int i = blockIdx.x * blockDim.x + threadIdx.x;
    if (i < total) {
        float di = dinv[i >> 7];
        out[i] = h[i] * di * di;
    }
}

// ---------------- edge scatter: one wave per edge, 4 feats/lane ----------------
__global__ __launch_bounds__(256) void edge_scatter_kernel(
        const int* __restrict__ ei, const float* __restrict__ ew,
        const float* __restrict__ dinv, const float* __restrict__ h,
        float* __restrict__ out, int E) {
    int e = blockIdx.x * 8 + (threadIdx.x >> 5);
    if (e >= E) return;
    const int lane = threadIdx.x & 31;
    const int s = ei[e];           // source (row)
    const int t = ei[E + e];       // target (col)
    const float nrm = dinv[s] * ew[e] * dinv[t];
    const float4 hv = *(const float4*)(h + (size_t)s * DIM + lane * 4);
    float* op = out + (size_t)t * DIM + lane * 4;
    atomicAdd(op + 0, hv.x * nrm);
    atomicAdd(op + 1, hv.y * nrm);
    atomicAdd(op + 2, hv.z * nrm);
    atomicAdd(op + 3, hv.w * nrm);
}

// ---------------- bias + exact-erf GELU, in place ----------------
__global__ void bias_gelu_kernel(float* __restrict__ out, const float* __restrict__ b, int total) {
    int i = blockIdx.x * blockDim.x + threadIdx.x;
    if (i < total) {
        float v = out[i] + b[i & (DIM - 1)];
        out[i] = 0.5f * v * (1.0f + erff(v * 0.70710678118654752440f));
    }
}

extern "C" void kernel_launch(void* const* d_in, const int* in_sizes, int n_in,
                              void* d_out, int out_size, void* d_ws, size_t ws_size,
                              hipStream_t stream) {
    const float* x   = (const float*)d_in[0];
    const int*   ei  = (const int*)  d_in[1];   // [2, E] flattened: rows then cols
    const float* ew  = (const float*)d_in[2];
    const float* W   = (const float*)d_in[3];
    const float* b   = (const float*)d_in[4];
    const float* lnw = (const float*)d_in[5];
    const float* lnb = (const float*)d_in[6];
    float* out = (float*)d_out;

    const int n = in_sizes[0] / DIM;
    const int E = in_sizes[1] / 2;
    const int total = n * DIM;

    // workspace layout (floats): deg[n] | dinv[n] | h[n*DIM]
    float* deg  = (float*)d_ws;
    float* dinv = deg + n;
    float* h    = dinv + n;

    init_deg_kernel   <<<(n + 255) / 256, 256, 0, stream>>>(deg, n);
    edge_deg_kernel   <<<(E + 255) / 256, 256, 0, stream>>>(ei, ew, deg, E);
    dinv_kernel       <<<(n + 255) / 256, 256, 0, stream>>>(deg, dinv, n);
    ln_gemm_wmma_kernel<<<(n + 15) / 16, 128, 0, stream>>>(x, W, lnw, lnb, h, n);
    selfloop_init_kernel<<<(total + 255) / 256, 256, 0, stream>>>(h, dinv, out, total);
    edge_scatter_kernel<<<(E + 7) / 8, 256, 0, stream>>>(ei, ew, dinv, h, out, E);
    bias_gelu_kernel  <<<(total + 255) / 256, 256, 0, stream>>>(out, b, total);
}